// GCN_8478265442663
// MI455X (gfx1250) — compile-verified
//
#include <hip/hip_runtime.h>

typedef __attribute__((ext_vector_type(2))) float v2f;
typedef __attribute__((ext_vector_type(8))) float v8f;

// ---------------------------------------------------------------- utilities

__global__ void fill_f32(float* __restrict__ p, float v, long long n) {
  long long i = (long long)blockIdx.x * blockDim.x + threadIdx.x;
  if (i < n) p[i] = v;
}

__global__ void deg_accum(const int* __restrict__ dst, float* __restrict__ deg, int E) {
  int e = blockIdx.x * blockDim.x + threadIdx.x;
  if (e < E) atomicAdd(&deg[dst[e]], 1.0f);
}

__global__ void rsqrt_inplace(float* __restrict__ p, int n) {
  int i = blockIdx.x * blockDim.x + threadIdx.x;
  if (i < n) p[i] = rsqrtf(p[i]);
}

// ------------------------------------------------- fp32 WMMA GEMM: HW = H*W
// One block: 16 output rows. wave w -> 16-col tile [16w, 16w+16).
// A tile (16 x K) staged in LDS; W streamed from global (L2-resident, 64 KB).
template<int K, int NOUT>
__global__ __launch_bounds__(32 * (NOUT / 16))
void gemm_wmma(const float* __restrict__ H, const float* __restrict__ W,
               float* __restrict__ HW, int nrows) {
  __shared__ float As[16 * K];
  const int lane = threadIdx.x & 31;
  const int wave = threadIdx.x >> 5;
  const int row0 = blockIdx.x * 16;
  const bool full = (row0 + 16 <= nrows);

  // cooperative float4 load of the contiguous 16xK row-major A tile
  const int nvec = (16 * K) / 4;
  if (full) {
    const float4* srcp = (const float4*)(H + (size_t)row0 * K);
    for (int i = threadIdx.x; i < nvec; i += blockDim.x)
      ((float4*)As)[i] = srcp[i];
  } else {
    for (int i = threadIdx.x; i < nvec; i += blockDim.x) {
      int r = i / (K / 4);
      float4 v = {0.f, 0.f, 0.f, 0.f};
      if (row0 + r < nrows)
        v = ((const float4*)(H + (size_t)row0 * K))[i];
      ((float4*)As)[i] = v;
    }
  }
  __syncthreads();

  // ISA 32-bit 16x4 A/B fragment striping:
  // lane L: M/N = L%16, K pair base = 2*(L/16)
  const int col0 = wave * 16;
  const int r  = lane & 15;
  const int kb = (lane >> 4) << 1;

  v8f acc = {};
#pragma unroll 4
  for (int k0 = 0; k0 < K; k0 += 4) {
    v2f a, b;
    a.x = As[r * K + k0 + kb];
    a.y = As[r * K + k0 + kb + 1];
    b.x = W[(size_t)(k0 + kb)     * NOUT + col0 + r];
    b.y = W[(size_t)(k0 + kb + 1) * NOUT + col0 + r];
    // D = A(16x4) * B(4x16) + C, fp32 throughout
    acc = __builtin_amdgcn_wmma_f32_16x16x4_f32(false, a, false, b,
                                                (short)0, acc, false, false);
  }

  // C/D layout: VGPR i -> M = i (lanes 0-15) / M = i+8 (lanes 16-31), N = L%16
  const int moff = (lane >> 4) * 8;
  const int col  = col0 + r;
  float* outp = HW + (size_t)(row0 + moff) * NOUT + col;
  if (full) {
    // 8 stores with constant immediate offsets -> single store clause
#pragma unroll
    for (int i = 0; i < 8; ++i)
      outp[(size_t)i * NOUT] = acc[i];
  } else {
#pragma unroll
    for (int i = 0; i < 8; ++i)
      if (row0 + moff + i < nrows) outp[(size_t)i * NOUT] = acc[i];
  }
}

// -------------------------------------- per-edge gather + normalized scatter
// one thread = 4 channels of one edge; for C=128 a wave32 covers one edge
// with a fully coalesced 512B gather; scatter via f32 atomics (L2-resident).
template<int C>
__global__ void edge_scatter(const float* __restrict__ HW, const int* __restrict__ src,
                             const int* __restrict__ dst, const float* __restrict__ dinv,
                             float* __restrict__ agg, long long total) {
  long long idx = (long long)blockIdx.x * blockDim.x + threadIdx.x;
  if (idx >= total) return;
  constexpr int CPE = C / 4;
  long long e = idx / CPE;
  int c4 = (int)(idx - e * CPE);
  int s = src[e];
  int d = dst[e];
  float norm = dinv[s] * dinv[d];
  const float4 m = *(const float4*)(HW + (size_t)s * C + (size_t)c4 * 4);
  float* o = agg + (size_t)d * C + (size_t)c4 * 4;
  atomicAdd(o + 0, m.x * norm);
  atomicAdd(o + 1, m.y * norm);
  atomicAdd(o + 2, m.z * norm);
  atomicAdd(o + 3, m.w * norm);
}

// ------------------------- epilogue: out = relu(agg + dinv^2 * hW + bias)
template<int C>
__global__ void node_final(const float* __restrict__ agg, const float* __restrict__ HW,
                           const float* __restrict__ dinv, const float* __restrict__ bias,
                           float* __restrict__ out, long long total) {
  long long idx = (long long)blockIdx.x * blockDim.x + threadIdx.x;
  if (idx >= total) return;  // total = N * C/4
  constexpr int CPE = C / 4;
  long long node = idx / CPE;
  int c4 = (int)(idx - node * CPE);
  float di = dinv[node];
  float sl = di * di;
  float4 a = ((const float4*)agg)[idx];
  float4 h = ((const float4*)HW)[idx];
  float4 b = *(const float4*)(bias + (size_t)c4 * 4);
  float4 r;
  r.x = fmaxf(a.x + sl * h.x + b.x, 0.f);
  r.y = fmaxf(a.y + sl * h.y + b.y, 0.f);
  r.z = fmaxf(a.z + sl * h.z + b.z, 0.f);
  r.w = fmaxf(a.w + sl * h.w + b.w, 0.f);
  ((float4*)out)[idx] = r;
}

// ---------------------------------------------------------------- launcher

static inline long long cdivll(long long a, long long b) { return (a + b - 1) / b; }

extern "C" void kernel_launch(void* const* d_in, const int* in_sizes, int n_in,
                              void* d_out, int out_size, void* d_ws, size_t ws_size,
                              hipStream_t stream) {
  const float* x  = (const float*)d_in[0];
  const int*   ei = (const int*)d_in[1];
  const float* W0 = (const float*)d_in[2];
  const float* b0 = (const float*)d_in[3];
  const float* W1 = (const float*)d_in[4];
  const float* b1 = (const float*)d_in[5];
  const float* W2 = (const float*)d_in[6];
  const float* b2 = (const float*)d_in[7];

  const int Nn = in_sizes[0] / 128;   // 100000
  const int Ee = in_sizes[1] / 2;     // 1600000
  const int* src = ei;
  const int* dst = ei + Ee;

  float* w = (float*)d_ws;
  size_t off = 0;
  float* dinv = w + off; off += ((size_t)Nn + 255) & ~(size_t)255;
  float* hw   = w + off; off += (size_t)Nn * 128;
  float* agg  = w + off; off += (size_t)Nn * 128;
  float* hbuf = w + off;

  const int T = 256;

  // degree (self-loop counts as 1 -> init 1.0), then dinv = rsqrt(deg)
  fill_f32<<<(unsigned)cdivll(Nn, T), T, 0, stream>>>(dinv, 1.0f, Nn);
  deg_accum<<<(unsigned)cdivll(Ee, T), T, 0, stream>>>(dst, dinv, Ee);
  rsqrt_inplace<<<(unsigned)cdivll(Nn, T), T, 0, stream>>>(dinv, Nn);

  const unsigned mblocks = (unsigned)((Nn + 15) / 16);

  // ---- layer 0: 128 -> 128
  gemm_wmma<128, 128><<<mblocks, 256, 0, stream>>>(x, W0, hw, Nn);
  fill_f32<<<(unsigned)cdivll((long long)Nn * 128, T), T, 0, stream>>>(agg, 0.f, (long long)Nn * 128);
  edge_scatter<128><<<(unsigned)cdivll((long long)Ee * 32, T), T, 0, stream>>>(
      hw, src, dst, dinv, agg, (long long)Ee * 32);
  node_final<128><<<(unsigned)cdivll((long long)Nn * 32, T), T, 0, stream>>>(
      agg, hw, dinv, b0, hbuf, (long long)Nn * 32);

  // ---- layer 1: 128 -> 128
  gemm_wmma<128, 128><<<mblocks, 256, 0, stream>>>(hbuf, W1, hw, Nn);
  fill_f32<<<(unsigned)cdivll((long long)Nn * 128, T), T, 0, stream>>>(agg, 0.f, (long long)Nn * 128);
  edge_scatter<128><<<(unsigned)cdivll((long long)Ee * 32, T), T, 0, stream>>>(
      hw, src, dst, dinv, agg, (long long)Ee * 32);
  node_final<128><<<(unsigned)cdivll((long long)Nn * 32, T), T, 0, stream>>>(
      agg, hw, dinv, b1, hbuf, (long long)Nn * 32);

  // ---- layer 2: 128 -> 64
  gemm_wmma<128, 64><<<mblocks, 128, 0, stream>>>(hbuf, W2, hw, Nn);
  fill_f32<<<(unsigned)cdivll((long long)Nn * 64, T), T, 0, stream>>>(agg, 0.f, (long long)Nn * 64);
  edge_scatter<64><<<(unsigned)cdivll((long long)Ee * 16, T), T, 0, stream>>>(
      hw, src, dst, dinv, agg, (long long)Ee * 16);
  node_final<64><<<(unsigned)cdivll((long long)Nn * 16, T), T, 0, stream>>>(
      agg, hw, dinv, b2, (float*)d_out, (long long)Nn * 16);
}